// BackboneModel_45655502357210
// MI455X (gfx1250) — compile-verified
//
#include <hip/hip_runtime.h>
#include <cstdint>

// ---------------------------------------------------------------------------
// BackboneModel frame-placement kernel for MI455X (gfx1250).
// Pure streaming op: 264 MB total traffic @ 23.3 TB/s => ~11.3 us floor.
// Strategy: tile-stage through LDS with gfx1250 async global<->LDS DMA so all
// global traffic is fully coalesced b128, then per-thread frame math in VGPRs.
// BLOCK=128, TILE=256 makes every loop trip count exact (21 DMA iters, 2
// residues/thread) => no divergent remainder code.
// ---------------------------------------------------------------------------

typedef float v4f __attribute__((ext_vector_type(4)));
typedef int v4i __attribute__((vector_size(16)));

// Address-space-qualified int4 pointers for the gfx1250 async DMA builtins
// (expected param type per clang: int4 in AS1 / AS3).
typedef __attribute__((address_space(1))) v4i gv4i;
typedef __attribute__((address_space(3))) v4i lv4i;

#define BLOCK 128
#define TILE 256                    // residues per block
#define FPR 42                      // floats per residue (14 atoms * 3)
#define TILE_FLOATS (TILE * FPR)    // 10752 floats = 42 KB LDS
#define TILE_VEC4 (TILE_FLOATS / 4) // 2688 float4s -> 21 iters of 128 threads

#if defined(__AMDGCN__) && __has_builtin(__builtin_amdgcn_global_load_async_to_lds_b128)
#define USE_ASYNC_LD 1
#else
#define USE_ASYNC_LD 0
#endif
#if defined(__AMDGCN__) && __has_builtin(__builtin_amdgcn_global_store_async_from_lds_b128)
#define USE_ASYNC_ST 1
#else
#define USE_ASYNC_ST 0
#endif

__device__ __forceinline__ void wait_asynccnt0() {
#if defined(__AMDGCN__)
#if __has_builtin(__builtin_amdgcn_s_wait_asynccnt)
  __builtin_amdgcn_s_wait_asynccnt(0);
#else
  asm volatile("s_wait_asynccnt 0" ::: "memory");
#endif
#endif
}

__device__ __forceinline__ void compute_residue(float* f) {
  const float Nx = f[0], Ny = f[1], Nz = f[2];
  const float Ax = f[3], Ay = f[4], Az = f[5];
  const float Cx = f[6], Cy = f[7], Cz = f[8];

  const float eps2 = 0.001f * 0.001f;

  // e1 = normalize(C - CA)
  const float d1x = Cx - Ax, d1y = Cy - Ay, d1z = Cz - Az;
  const float inv1 = rsqrtf(d1x * d1x + d1y * d1y + d1z * d1z + eps2);
  const float e1x = d1x * inv1, e1y = d1y * inv1, e1z = d1z * inv1;

  // e2 = normalize((N - CA) - ((N-CA).e1) e1)
  const float vx = Nx - Ax, vy = Ny - Ay, vz = Nz - Az;
  const float p = vx * e1x + vy * e1y + vz * e1z;
  const float wx = vx - p * e1x, wy = vy - p * e1y, wz = vz - p * e1z;
  const float inv2 = rsqrtf(wx * wx + wy * wy + wz * wz + eps2);
  const float e2x = wx * inv2, e2y = wy * inv2, e2z = wz * inv2;

  // X_IDEAL rows (z column is zero, e3 never contributes):
  //   a0: (-0.525, 1.363)  a1: (0,0)  a2: (1.526, 0)  a3: (2.153, -1.062)
  const float c00 = -0.525f, c01 = 1.363f;
  const float c20 = 1.526f;
  const float c30 = 2.153f, c31 = -1.062f;

  f[0] = fmaf(c00, e1x, fmaf(c01, e2x, Ax));
  f[1] = fmaf(c00, e1y, fmaf(c01, e2y, Ay));
  f[2] = fmaf(c00, e1z, fmaf(c01, e2z, Az));
  f[3] = Ax;
  f[4] = Ay;
  f[5] = Az;
  f[6] = fmaf(c20, e1x, Ax);
  f[7] = fmaf(c20, e1y, Ay);
  f[8] = fmaf(c20, e1z, Az);
  f[9] = fmaf(c30, e1x, fmaf(c31, e2x, Ax));
  f[10] = fmaf(c30, e1y, fmaf(c31, e2y, Ay));
  f[11] = fmaf(c30, e1z, fmaf(c31, e2z, Az));
  // floats 12..41 (atoms 4..13) pass through untouched.
}

__global__ __launch_bounds__(BLOCK) void bb_frames_kernel(
    const float* __restrict__ X, float* __restrict__ out, int n) {
  __shared__ __align__(16) float s[TILE_FLOATS];

  const int tid = threadIdx.x;
  const long long base = (long long)blockIdx.x * TILE;
  const long long rem = (long long)n - base;
  if (rem <= 0) return;
  const int nres = rem < TILE ? (int)rem : TILE;

  const float* __restrict__ gsrc = X + base * FPR;
  float* __restrict__ gdst = out + base * FPR;

  // ---- Phase 1: tile global -> LDS (coalesced b128 async DMA, 21 exact iters)
  if (nres == TILE) {
#pragma unroll
    for (int i = tid; i < TILE_VEC4; i += BLOCK) {
#if USE_ASYNC_LD
      __builtin_amdgcn_global_load_async_to_lds_b128(
          (gv4i*)(void*)(gsrc + 4 * i), (lv4i*)(void*)(s + 4 * i), 0, 0);
#else
      ((v4f*)s)[i] = __builtin_nontemporal_load((const v4f*)gsrc + i);
#endif
    }
#if USE_ASYNC_LD
    wait_asynccnt0();
#endif
    // Prefetch the next tile while we compute (gfx1250 global_prefetch_b8).
    // 128 threads * 336 B = 43008 B = exactly one tile.
    if (base + TILE < (long long)n) {
      __builtin_prefetch((const char*)(gsrc + TILE_FLOATS) + tid * 336, 0, 0);
    }
  } else {
    const int nf = nres * FPR;
    for (int i = tid; i < nf; i += BLOCK) s[i] = gsrc[i];
  }
  __syncthreads();

  // ---- Phase 2: frame math, 2 residues per thread; overwrite atoms 0..3.
  for (int r = tid; r < nres; r += BLOCK) {
    compute_residue(s + r * FPR);
  }
  __syncthreads();

  // ---- Phase 3: tile LDS -> global (coalesced b128 async DMA).
  // ASYNCcnt drains at the implicit wait-idle before s_endpgm.
  if (nres == TILE) {
#pragma unroll
    for (int i = tid; i < TILE_VEC4; i += BLOCK) {
#if USE_ASYNC_ST
      __builtin_amdgcn_global_store_async_from_lds_b128(
          (gv4i*)(void*)(gdst + 4 * i), (lv4i*)(void*)(s + 4 * i), 0, 0);
#else
      __builtin_nontemporal_store(((const v4f*)s)[i], (v4f*)gdst + i);
#endif
    }
  } else {
    const int nf = nres * FPR;
    for (int i = tid; i < nf; i += BLOCK) gdst[i] = s[i];
  }
}

extern "C" void kernel_launch(void* const* d_in, const int* in_sizes, int n_in,
                              void* d_out, int out_size, void* d_ws, size_t ws_size,
                              hipStream_t stream) {
  (void)d_ws;
  (void)ws_size;
  (void)n_in;
  (void)out_size;
  const float* X = (const float*)d_in[0];
  // batch_ids / max_len are identity plumbing in the reference; only n matters.
  const int n = in_sizes[1];  // number of residues (batch_ids element count)
  float* out = (float*)d_out;

  const int blocks = (n + TILE - 1) / TILE;
  bb_frames_kernel<<<blocks, BLOCK, 0, stream>>>(X, out, n);
}